// MoELoRALayer_31026843746610
// MI455X (gfx1250) — compile-verified
//
#include <hip/hip_runtime.h>
#include <hip/hip_bf16.h>

typedef __attribute__((ext_vector_type(2))) float v2f;
typedef __attribute__((ext_vector_type(8))) float v8f;
typedef __attribute__((ext_vector_type(4))) int   v4i;

#define T_TOKENS 8192
#define DIM      4096   // D_IN == D_OUT
#define N_EXP    8
#define RANK     16
#define KLORA    128    // E * R
#define LORA_SCALE 2.0f // 32 / 16

// ---------------- generic WMMA-f32 GEMM:  C[M,N] (=|+=) A[M,K] * W[N,K]^T (+bias)
#define BM 64
#define BN 128
#define KC 32
#define LDSS 36   // padded LDS row stride (floats): conflict-free, 16B-aligned

#if __has_builtin(__builtin_amdgcn_global_load_async_to_lds_b128)
#define HAS_ASYNC_LDS 1
#define GLOBAL_AS __attribute__((address_space(1)))
#define LDS_AS    __attribute__((address_space(3)))
#else
#define HAS_ASYNC_LDS 0
#endif

__device__ __forceinline__ void async_copy_b128(const float* g, float* l)
{
#if HAS_ASYNC_LDS
    __builtin_amdgcn_global_load_async_to_lds_b128(
        (GLOBAL_AS v4i*)g, (LDS_AS v4i*)l, 0, 0);
#else
    *(float4*)l = *(const float4*)g;
#endif
}

__device__ __forceinline__ void wait_async0()
{
#if HAS_ASYNC_LDS
#if __has_builtin(__builtin_amdgcn_s_wait_asynccnt)
    __builtin_amdgcn_s_wait_asynccnt(0);
#else
    asm volatile("s_wait_asynccnt 0x0" ::: "memory");
#endif
#endif
}

template <bool BIAS, bool ACCUM>
__global__ __launch_bounds__(256, 2)
void gemm_wmma_f32(const float* __restrict__ A, const float* __restrict__ W,
                   const float* __restrict__ bias, float* __restrict__ C,
                   int M, int N, int K)
{
    __shared__ float lx[2][BM * LDSS];   // 2 x  9,216 B
    __shared__ float lw[2][BN * LDSS];   // 2 x 18,432 B

    const int tid   = threadIdx.x;
    const int lane  = tid & 31;
    const int wave  = tid >> 5;        // 8 waves: 2 (M) x 4 (N)
    const int wm    = wave >> 2;       // 0..1
    const int wn    = wave & 3;        // 0..3
    const int lhalf = lane >> 4;       // 0..1
    const int lmod  = lane & 15;       // 0..15
    const int m0 = blockIdx.y * BM;
    const int n0 = blockIdx.x * BN;

    v8f acc[2][2];
    {
        v8f z = {};
        acc[0][0] = z; acc[0][1] = z; acc[1][0] = z; acc[1][1] = z;
    }

    // staging assignments (256 threads), all 16B-aligned on both sides
    const int ax_row = tid >> 2;            // 0..63
    const int ax_col = (tid & 3) * 8;       // 0,8,16,24  (2 x b128)
    const int aw_row = tid >> 1;            // 0..127
    const int aw_col = (tid & 1) * 16;      // 0,16       (4 x b128)

    const float* agp = A + (size_t)(m0 + ax_row) * K + ax_col;
    const float* wgp = W + (size_t)(n0 + aw_row) * K + aw_col;
    float* const axl = &lx[0][0] + ax_row * LDSS + ax_col;
    float* const awl = &lw[0][0] + aw_row * LDSS + aw_col;
    const int lxsz = BM * LDSS;
    const int lwsz = BN * LDSS;

    // stage K-tile `ko` into buffer `buf`
    auto stage = [&](int buf, int ko) {
        const float* ga = agp + ko;
        float* la = axl + buf * lxsz;
        async_copy_b128(ga,     la);
        async_copy_b128(ga + 4, la + 4);
        const float* gw = wgp + ko;
        float* lwp = awl + buf * lwsz;
        async_copy_b128(gw,      lwp);
        async_copy_b128(gw + 4,  lwp + 4);
        async_copy_b128(gw + 8,  lwp + 8);
        async_copy_b128(gw + 12, lwp + 12);
    };

    stage(0, 0);                 // prologue: fill buffer 0
    int cur = 0;

    for (int ko = 0; ko < K; ko += KC) {
        wait_async0();           // my async copies into buf[cur] are done
        __syncthreads();         // everyone's copies (and all reads of buf[cur^1]) done

        if (ko + KC < K)
            stage(cur ^ 1, ko + KC);   // overlap next tile's DMA with compute

        const float* lxc = &lx[cur][0];
        const float* lwc = &lw[cur][0];
        #pragma unroll
        for (int kk = 0; kk < KC; kk += 4) {
            const int c = kk + 2 * lhalf;  // even -> 8B-aligned ds_load_b64
            v2f a[2], b[2];
            #pragma unroll
            for (int i = 0; i < 2; ++i)
                a[i] = *(const v2f*)(lxc + (wm * 32 + i * 16 + lmod) * LDSS + c);
            #pragma unroll
            for (int j = 0; j < 2; ++j)
                b[j] = *(const v2f*)(lwc + (wn * 32 + j * 16 + lmod) * LDSS + c);
            #pragma unroll
            for (int i = 0; i < 2; ++i)
                #pragma unroll
                for (int j = 0; j < 2; ++j)
                    acc[i][j] = __builtin_amdgcn_wmma_f32_16x16x4_f32(
                        false, a[i], false, b[j], (short)0, acc[i][j], false, false);
        }
        cur ^= 1;
    }

    // epilogue: C/D layout -> m = vgpr + 8*lhalf, n = lmod (per 16x16 tile)
    #pragma unroll
    for (int j = 0; j < 2; ++j) {
        const int n = n0 + wn * 32 + j * 16 + lmod;
        const float bval = BIAS ? bias[n] : 0.0f;
        #pragma unroll
        for (int i = 0; i < 2; ++i) {
            #pragma unroll
            for (int v = 0; v < 8; ++v) {
                const int m = m0 + wm * 32 + i * 16 + v + 8 * lhalf;
                float* p = C + (size_t)m * N + n;
                float r = acc[i][j][v] + bval;
                if (ACCUM) r += *p;
                *p = r;
            }
        }
    }
}

// ---------------- router: one wave per token; top-2 softmax weights, dense [T,8]
__global__ __launch_bounds__(256)
void router_kernel(const float* __restrict__ x, const float* __restrict__ rw,
                   float* __restrict__ wout)
{
    const int lane = threadIdx.x & 31;
    const int wave = threadIdx.x >> 5;
    const int t = blockIdx.x * 8 + wave;

    float accv[N_EXP];
    #pragma unroll
    for (int e = 0; e < N_EXP; ++e) accv[e] = 0.0f;

    const float* xp = x + (size_t)t * DIM;
    for (int d = lane; d < DIM; d += 32) {
        const float xv = xp[d];
        #pragma unroll
        for (int e = 0; e < N_EXP; ++e)
            accv[e] += xv * rw[e * DIM + d];
    }
    #pragma unroll
    for (int e = 0; e < N_EXP; ++e) {
        #pragma unroll
        for (int off = 16; off > 0; off >>= 1)
            accv[e] += __shfl_xor(accv[e], off, 32);
    }
    if (lane == 0) {
        int i1 = 0; float m1 = accv[0];
        #pragma unroll
        for (int e = 1; e < N_EXP; ++e)
            if (accv[e] > m1) { m1 = accv[e]; i1 = e; }
        int i2 = (i1 == 0) ? 1 : 0; float m2 = accv[i2];
        #pragma unroll
        for (int e = 0; e < N_EXP; ++e)
            if (e != i1 && accv[e] > m2) { m2 = accv[e]; i2 = e; }
        // renormalized top-2 softmax (global denominator cancels)
        const float p1 = 1.0f / (1.0f + __expf(m2 - m1));
        const float p2 = 1.0f - p1;
        float* wp = wout + (size_t)t * N_EXP;
        #pragma unroll
        for (int e = 0; e < N_EXP; ++e) wp[e] = 0.0f;
        wp[i1] = p1;
        wp[i2] = p2;
    }
}

// ---------------- yw[t, e*16+r] = SCALE * w[t,e] * xA[t, e*16+r]
__global__ __launch_bounds__(256)
void scale_kernel(const float* __restrict__ xA, const float* __restrict__ w,
                  float* __restrict__ yw)
{
    const int idx = blockIdx.x * 256 + threadIdx.x;   // < T * 128
    const int t = idx >> 7;
    const int k = idx & 127;
    const int e = k >> 4;
    yw[idx] = LORA_SCALE * w[t * N_EXP + e] * xA[idx];
}

// ---------------- Wl[n, e*16+r] = lora_B[e, n, r]  (so GEMM loader sees [N,K] row-major)
__global__ __launch_bounds__(256)
void transpose_loraB_kernel(const float* __restrict__ loraB, float* __restrict__ Wl)
{
    const int idx = blockIdx.x * 256 + threadIdx.x;   // < DIM * 128
    const int n = idx >> 7;
    const int k = idx & 127;
    const int e = k >> 4;
    const int r = k & 15;
    Wl[idx] = loraB[(size_t)e * DIM * RANK + (size_t)n * RANK + r];
}

extern "C" void kernel_launch(void* const* d_in, const int* in_sizes, int n_in,
                              void* d_out, int out_size, void* d_ws, size_t ws_size,
                              hipStream_t stream)
{
    const float* x       = (const float*)d_in[0];  // [T, DIM]
    const float* W_base  = (const float*)d_in[1];  // [DIM, DIM]
    const float* b_base  = (const float*)d_in[2];  // [DIM]
    const float* routerw = (const float*)d_in[3];  // [E, DIM]
    const float* lora_A  = (const float*)d_in[4];  // [E, R, DIM] == [128, DIM] row-major
    const float* lora_B  = (const float*)d_in[5];  // [E, DIM, R]
    float* out = (float*)d_out;                    // [T, DIM]

    // workspace layout (floats): ~10.3 MB total
    float* ws  = (float*)d_ws;
    float* wrt = ws;                                // T * 8
    float* xA  = wrt + (size_t)T_TOKENS * N_EXP;    // T * 128
    float* yw  = xA  + (size_t)T_TOKENS * KLORA;    // T * 128
    float* Wl  = yw  + (size_t)T_TOKENS * KLORA;    // DIM * 128

    // 1) router weights
    router_kernel<<<T_TOKENS / 8, 256, 0, stream>>>(x, routerw, wrt);

    // 2) lora_B relayout (independent)
    transpose_loraB_kernel<<<(DIM * KLORA) / 256, 256, 0, stream>>>(lora_B, Wl);

    // 3) base GEMM: out = x @ W_base^T + b_base
    {
        dim3 grid(DIM / BN, T_TOKENS / BM);
        gemm_wmma_f32<true, false><<<grid, 256, 0, stream>>>(
            x, W_base, b_base, out, T_TOKENS, DIM, DIM);
    }

    // 4) xA = x @ A_flat^T   (N = 128)
    {
        dim3 grid(KLORA / BN, T_TOKENS / BM);
        gemm_wmma_f32<false, false><<<grid, 256, 0, stream>>>(
            x, lora_A, nullptr, xA, T_TOKENS, KLORA, DIM);
    }

    // 5) yw = SCALE * w * xA
    scale_kernel<<<(T_TOKENS * KLORA) / 256, 256, 0, stream>>>(xA, wrt, yw);

    // 6) out += yw @ Wl^T    (K = 128, accumulate)
    {
        dim3 grid(DIM / BN, T_TOKENS / BM);
        gemm_wmma_f32<false, true><<<grid, 256, 0, stream>>>(
            yw, Wl, nullptr, out, T_TOKENS, DIM, KLORA);
    }
}